// Lintul5Model_86199993631092
// MI455X (gfx1250) — compile-verified
//
#include <hip/hip_runtime.h>
#include <cstdint>

#define RADF 0.017453292519943295f
#define PIF  3.14159265358979323846f

#define T_STEPS 365
#define CHUNK   15          // steps per TDM chunk (45 dwords/row, odd -> conflict-free LDS)
#define TAIL    5           // 365 = 24*15 + 5 (15 dwords/row, also odd)
#define NCHUNK  25
#define ROWS    128         // threads (batch rows) per block

typedef unsigned int u32x4 __attribute__((ext_vector_type(4)));
typedef int          i32x8 __attribute__((ext_vector_type(8)));
typedef int          i32x4 __attribute__((ext_vector_type(4)));

__device__ __forceinline__ float clamp01(float x) { return fminf(fmaxf(x, 0.f), 1.f); }

// ---------------------------------------------------------------------------
// Kernel 1: per-day photoperiod factor / tsum_anth  (batch-independent astro)
// ---------------------------------------------------------------------------
__global__ void pfac_kernel(const float* __restrict__ cp,
                            const float* __restrict__ lat,
                            const int*   __restrict__ sdoy,
                            float*       __restrict__ A) {
    int t = blockIdx.x * blockDim.x + threadIdx.x;
    if (t >= T_STEPS) return;
    float doy   = (float)(((sdoy[0] - 1 + t) % 365) + 1);
    float dec   = -asinf(sinf(23.45f * RADF) * cosf(2.0f * PIF * (doy + 10.0f) / 365.0f));
    float latr  = lat[0] * RADF;
    float sinld = sinf(latr) * sinf(dec);
    float cosld = cosf(latr) * cosf(dec);
    float r     = fminf(fmaxf(sinld / cosld, -1.f), 1.f);
    float ddlp  = 12.f * (1.f + (2.f / PIF) * asinf(r));
    float pfac  = clamp01((ddlp - 8.f) / 6.f);
    A[t] = pfac / cp[4];   // pfac / tsum_anth
}

// ---------------------------------------------------------------------------
// TDM issue: 2D tile (tile_d0 elems/row, ROWS rows, row stride strideElems),
// data_size = 4B, dense pack into LDS at ldsAddr.  (D# layout per ISA ch.8)
// ---------------------------------------------------------------------------
__device__ __forceinline__ void tdm_issue(const void* gptr, unsigned ldsAddr,
                                          int tile_d0, int strideElems) {
    u32x4 g0; i32x8 g1; i32x4 g2, g3;
    unsigned long long ga = (unsigned long long)(uintptr_t)gptr;
    g0.x = 1u;                                                      // count=1
    g0.y = ldsAddr;                                                 // lds_addr
    g0.z = (unsigned)ga;                                            // global_addr lo
    g0.w = ((unsigned)(ga >> 32) & 0x01FFFFFFu) | (2u << 30);       // global_addr hi | type=2
    g1[0] = (int)(2u << 16);                                        // data_size=4B, mask=0
    g1[1] = (int)(((unsigned)strideElems & 0xFFFFu) << 16);         // tensor_dim0 lo
    g1[2] = (int)((((unsigned)strideElems >> 16) & 0xFFFFu)
                  | ((unsigned)ROWS << 16));                        // tensor_dim0 hi | tensor_dim1 lo
    g1[3] = (int)(((unsigned)tile_d0 & 0xFFFFu) << 16);             // tensor_dim1 hi(0) | tile_dim0
    g1[4] = ROWS;                                                   // tile_dim1 | tile_dim2=0
    g1[5] = strideElems;                                            // tensor_dim0_stride lo32
    g1[6] = 0;
    g1[7] = 0;
    g2[0] = 1; g2[1] = 1; g2[2] = 0; g2[3] = 0;                     // tensor_dim2=1, tensor_dim3=1
    g3[0] = 0; g3[1] = (1 << 16); g3[2] = 0; g3[3] = 0;             // tensor_dim4=1
#if __has_include(<hip/amd_detail/amd_gfx1250_TDM.h>)
    i32x8 gz = {};
    __builtin_amdgcn_tensor_load_to_lds(g0, g1, g2, g3, gz, 0);
#else
    __builtin_amdgcn_tensor_load_to_lds(g0, g1, g2, g3, 0);
#endif
}

// ---------------------------------------------------------------------------
// Kernel 2: one thread per batch element, TDM double-buffered weather staging
// ---------------------------------------------------------------------------
__global__ void __launch_bounds__(ROWS) lintul_kernel(
        const float* __restrict__ weather,   // [B, T, 3]
        const float* __restrict__ cp,        // [12]
        const float* __restrict__ sp,        // [4]
        const float* __restrict__ A,         // [365] pfac/tsum_an
        float* __restrict__ out, int B) {
    __shared__ float wbuf[2][ROWS * CHUNK * 3];
    __shared__ float Apf[T_STEPS];

    const int tid = threadIdx.x;
    const int b0  = blockIdx.x * ROWS;
    const int b   = b0 + tid;

    // uniform parameters (scalar loads)
    const float lue = cp[0], sla = cp[1], kk = cp[2], tbase = cp[3];
    const float tsum_an = cp[4], tsum_mat = cp[5], laii = cp[6], rootdi = cp[7];
    const float rootd_max = cp[8], rrdmax = cp[9], rdrl = cp[10], wccr_frac = cp[11];
    const float wcad = sp[0], wcwp = sp[1], wcfc = sp[2];
    const float inv_post = 1.0f / fmaxf(tsum_mat - tsum_an, 1.0f);
    const float wccr     = wcwp + wccr_frac * (wcfc - wcwp);
    const float inv_wr   = 1.0f / fmaxf(wccr - wcwp, 1e-6f);
    const float inv_ea   = 1.0f / fmaxf(wcfc - wcad, 1e-6f);

    // live state (tsum/wlvd/wrt are dead w.r.t. outputs)
    float dvs = 0.f, lai = laii, wlv = laii / fmaxf(sla, 1e-6f);
    float wst = 0.f, wso = 0.f, rootd = rootdi, wa = 1000.f * wcfc * rootdi;

    // output layout: yield[B] | lai[B,366] | dvs[B,366] | bio[B,366]
    float* laiT = out + B;
    float* dvsT = laiT + (size_t)B * (T_STEPS + 1);
    float* bioT = dvsT + (size_t)B * (T_STEPS + 1);
    const size_t obase = (size_t)b * (T_STEPS + 1);
    laiT[obase] = lai; dvsT[obase] = 0.f; bioT[obase] = wlv;

    const unsigned lds0 = (unsigned)(uintptr_t)&wbuf[0][0];
    const unsigned lds1 = (unsigned)(uintptr_t)&wbuf[1][0];
    const float* gbase  = weather + (size_t)b0 * T_STEPS * 3;

    if (tid < 32) tdm_issue(gbase, lds0, CHUNK * 3, T_STEPS * 3);   // wave 0 issues chunk 0
    for (int i = tid; i < T_STEPS; i += ROWS) Apf[i] = A[i];

    // one simulated day: rates from OLD state, Euler update, record trajectory
    auto step = [&](float dav, float irr, float rai, float apf, int t1) {
        const float teff  = fmaxf(dav - tbase, 0.f);
        const float dvsr  = (dvs < 1.f) ? teff * apf : teff * inv_post;
        const float ptot  = fmaxf(0.0075f * irr * (dav + 20.f), 0.f);
        const float fint  = 1.f - __expf(-kk * lai);
        const float pevap = ptot * (1.f - fint);
        const float ptran = ptot * fint;
        const float wc    = 0.001f * wa * __builtin_amdgcn_rcpf(fmaxf(rootd, 1e-3f));
        const float tranrf = clamp01((wc - wcwp) * inv_wr);
        const float tran  = ptran * tranrf;
        const float evap  = pevap * clamp01((wc - wcad) * inv_ea);
        const float can_grow = ((dvs < 1.f) && (rootd < rootd_max)) ? 1.f : 0.f;
        const float rootdr = rrdmax * can_grow * fminf(4.f * tranrf, 1.f);
        const float explor = 1000.f * wcfc * rootdr;
        const float drain  = fmaxf(wa + rai - tran - evap - 1000.f * wcfc * rootd, 0.f);
        const float war    = rai + explor - tran - evap - drain;
        const float parint = 0.5f * irr * fint;
        const float tfac   = clamp01(dav * 0.1f);
        const float gtot   = lue * parint * tfac * tranrf;
        const float frt    = fminf(fmaxf(0.5f * (1.f - dvs), 0.f), 0.5f);
        const float fso    = clamp01(dvs - 1.f);
        const float fsh    = 1.f - frt;
        const float gveg   = gtot * fsh * (1.f - fso);
        const float g_lv   = 0.6f * gveg;
        const float g_st   = 0.4f * gveg;
        const float g_so   = gtot * fsh * fso;
        const float rdr    = rdrl * fso;
        const float act    = (dvs < 2.f) ? 1.f : 0.f;

        const float wlv_old = wlv;
        wlv += act * (g_lv - wlv_old * rdr);
        wst += act * g_st;
        wso += act * g_so;
        lai  = fmaxf(lai + act * (sla * g_lv - lai * rdr), 0.f);
        dvs += dvsr;
        rootd += rootdr;
        wa = fmaxf(wa + war, 0.f);

        const size_t oi = obase + (size_t)t1;
        laiT[oi] = lai;
        dvsT[oi] = dvs;
        bioT[oi] = wlv + wst + wso;
    };

    for (int c = 0; c < NCHUNK; ++c) {
        if (tid < 32) {                                 // wave-uniform branch: wave 0 only
            if (c + 1 < NCHUNK) {
                const int ns = (c + 1 == NCHUNK - 1) ? TAIL : CHUNK;
                tdm_issue(gbase + (size_t)(c + 1) * CHUNK * 3,
                          ((c + 1) & 1) ? lds1 : lds0, ns * 3, T_STEPS * 3);
                __builtin_amdgcn_s_wait_tensorcnt(1);   // chunk c landed (in-order TDM)
            } else {
                __builtin_amdgcn_s_wait_tensorcnt(0);
            }
        }
        __syncthreads();

        if (c < NCHUNK - 1) {
            const float* wrow = &wbuf[c & 1][tid * (CHUNK * 3)];
            const int tbase0 = c * CHUNK;
#pragma unroll
            for (int s = 0; s < CHUNK; ++s)
                step(wrow[3 * s], wrow[3 * s + 1], wrow[3 * s + 2],
                     Apf[tbase0 + s], tbase0 + s + 1);
        } else {
            const float* wrow = &wbuf[c & 1][tid * (TAIL * 3)];
            const int tbase0 = c * CHUNK;
#pragma unroll
            for (int s = 0; s < TAIL; ++s)
                step(wrow[3 * s], wrow[3 * s + 1], wrow[3 * s + 2],
                     Apf[tbase0 + s], tbase0 + s + 1);
        }
        __syncthreads();   // buf[c&1] free for chunk c+2's TDM
    }
    out[b] = wso;          // yield = final WSO
}

// ---------------------------------------------------------------------------
extern "C" void kernel_launch(void* const* d_in, const int* in_sizes, int n_in,
                              void* d_out, int out_size, void* d_ws, size_t ws_size,
                              hipStream_t stream) {
    (void)n_in; (void)out_size; (void)ws_size;
    const float* weather = (const float*)d_in[0];
    const float* cp      = (const float*)d_in[1];
    const float* sp      = (const float*)d_in[2];
    const float* lat     = (const float*)d_in[3];
    const int*   sdoy    = (const int*)d_in[4];
    float* out = (float*)d_out;
    float* A   = (float*)d_ws;                 // 365 floats of scratch

    const int B = in_sizes[0] / (T_STEPS * 3); // 16384

    pfac_kernel<<<1, 384, 0, stream>>>(cp, lat, sdoy, A);
    lintul_kernel<<<B / ROWS, ROWS, 0, stream>>>(weather, cp, sp, A, out, B);
}